// KeyHeatModule_test2_47682726920693
// MI455X (gfx1250) — compile-verified
//
#include <hip/hip_runtime.h>
#include <hip/hip_bf16.h>

typedef __attribute__((ext_vector_type(2))) float v2f;
typedef __attribute__((ext_vector_type(8))) float v8f;
typedef __attribute__((ext_vector_type(2))) int   v2i;

#define AS1 __attribute__((address_space(1)))
#define AS3 __attribute__((address_space(3)))
typedef AS1 const float* gcf;   // global const float*
typedef AS1 float*       gf;    // global float*
typedef AS1 const v2f*   gcv2f; // global const float2*

#define K_DIM 8330
#define BSZ 4096
#define NCOL 126
#define NCOL_PAD 128
#define KC 32
#define BROWS 64
#define WD 7

#if __has_builtin(__builtin_amdgcn_global_load_async_to_lds_b64)
#define USE_ASYNC_LDS 1
#else
#define USE_ASYNC_LDS 0
#endif

__device__ __forceinline__ void wait_async0() {
#if __has_builtin(__builtin_amdgcn_s_wait_asynccnt)
  __builtin_amdgcn_s_wait_asynccnt(0);
#else
  asm volatile("s_wait_asynccnt 0x0" ::: "memory");
#endif
}

// ---------------------------------------------------------------------------
// GEMM: vec = x @ [W_head|W_arm|W_upper|W_lower|W_foot] + bias  (f32 WMMA)
// ---------------------------------------------------------------------------
__device__ __forceinline__ v8f wmma_f32(v2f a, v2f b, v8f c) {
  // 8 args: (neg_a, A, neg_b, B, c_mod, C, reuse_a, reuse_b)
  return __builtin_amdgcn_wmma_f32_16x16x4_f32(
      false, a, false, b, (short)0, c, false, false);
}

__global__ __launch_bounds__(128) void gemm_kernel(
    const float* __restrict__ x,
    const float* __restrict__ Wh, const float* __restrict__ bh,
    const float* __restrict__ Wa, const float* __restrict__ ba,
    const float* __restrict__ Wu, const float* __restrict__ bu,
    const float* __restrict__ Wl, const float* __restrict__ bl,
    const float* __restrict__ Wf, const float* __restrict__ bf,
    float* __restrict__ out)
{
  // 64 rows x 32 K-floats per stage, stored as float2 with pitch 17 (34 floats)
  // -> conflict-free for the A-fragment read pattern of V_WMMA_F32_16X16X4_F32.
  __shared__ v2f A2[BROWS][17];

  const int tid   = threadIdx.x;
  const int lane  = tid & 31;
  const int wave  = tid >> 5;
  const int rowbase = blockIdx.x * BROWS;
  const int nlo   = lane & 15;     // N index within a 16-wide tile
  const int khalf = lane >> 4;     // 0: K={0,1}, 1: K={2,3} within a K=4 step
  const int koff  = khalf * 2;

  gcf xg = (gcf)x;
  gf  og = (gf)out;
  gcf Wp[5] = {(gcf)Wh, (gcf)Wa, (gcf)Wu, (gcf)Wl, (gcf)Wf};
  gcf Bp[5] = {(gcf)bh, (gcf)ba, (gcf)bu, (gcf)bl, (gcf)bf};

  // Per-lane column -> (weight block, local col, stride, bias)
  gcf   wptr[2];
  int   wstr[2];
  float bias[2];
  bool  valid[2];
  for (int t = 0; t < 2; ++t) {
    int col = (wave * 2 + t) * 16 + nlo;
    bool v  = col < NCOL;
    int  c  = v ? col : 0;
    int  p  = (c < 28) ? 0 : (c < 56) ? 1 : (c < 84) ? 2 : (c < 112) ? 3 : 4;
    int  off = (p == 4) ? 112 : p * 28;
    int  local = c - off;
    wptr[t]  = Wp[p] + local;
    wstr[t]  = (p == 4) ? 14 : 28;
    bias[t]  = v ? Bp[p][local] : 0.0f;
    valid[t] = v;
  }

  // Accumulators: 4 M-tiles x 2 N-tiles, bias folded into C init
  v8f acc[4][2];
  for (int mt = 0; mt < 4; ++mt)
    for (int t = 0; t < 2; ++t)
      for (int i = 0; i < 8; ++i)
        acc[mt][t][i] = bias[t];

  for (int kc = 0; kc < K_DIM; kc += KC) {
    __syncthreads();
    // Cooperative coalesced stage of x[rowbase:rowbase+64, kc:kc+32] into LDS.
    #pragma unroll
    for (int j = 0; j < 8; ++j) {
      int e  = tid + 128 * j;          // 0..1023 float2 elements
      int r  = e >> 4;                 // row within block tile
      int k2 = e & 15;                 // float2 index within the 32-wide chunk
      int kk = kc + k2 * 2;
      if (kk < K_DIM) {                // K_DIM even -> kk+1 also valid
#if USE_ASYNC_LDS
        // CDNA5 async Global->LDS copy: no VGPR round-trip, tracked by ASYNCcnt
        __builtin_amdgcn_global_load_async_to_lds_b64(
            (AS1 v2i*)(xg + (size_t)(rowbase + r) * K_DIM + kk),
            (AS3 v2i*)&A2[r][k2], 0, 0);
#else
        A2[r][k2] = *(gcv2f)(xg + (size_t)(rowbase + r) * K_DIM + kk);
#endif
      } else {
        v2f z; z.x = 0.0f; z.y = 0.0f; // zero-pad tail: products vanish
        A2[r][k2] = z;
      }
    }
    if (kc + KC < K_DIM)               // hint next chunk -> global_prefetch_b8
      __builtin_prefetch(
          x + (size_t)(rowbase + (tid >> 4)) * K_DIM + (kc + KC) + (tid & 15) * 2,
          0, 0);
#if USE_ASYNC_LDS
    wait_async0();
#endif
    __syncthreads();

    #pragma unroll
    for (int ks = 0; ks < KC; ks += 4) {
      int kg  = kc + ks + koff;
      int kcl = (kg <= K_DIM - 2) ? kg : (K_DIM - 2);  // safe addr; A is zeroed
      // B fragments: B[k][n], B[k+1][n] for each of the wave's two N-tiles.
      v2f bv[2];
      #pragma unroll
      for (int t = 0; t < 2; ++t) {
        v2f bb;
        bb.x = wptr[t][(size_t)kcl * wstr[t]];
        bb.y = wptr[t][(size_t)(kcl + 1) * wstr[t]];
        bv[t] = bb;
      }
      // A fragments from LDS, reused across both N-tiles.
      #pragma unroll
      for (int mt = 0; mt < 4; ++mt) {
        v2f av = A2[mt * 16 + nlo][(ks >> 1) + khalf];
        acc[mt][0] = wmma_f32(av, bv[0], acc[mt][0]);
        acc[mt][1] = wmma_f32(av, bv[1], acc[mt][1]);
      }
    }
  }

  // Store D: VGPR i, lanes 0-15 -> M=i, lanes 16-31 -> M=i+8 ; N = lane&15.
  for (int mt = 0; mt < 4; ++mt)
    for (int t = 0; t < 2; ++t) {
      if (!valid[t]) continue;
      int c = (wave * 2 + t) * 16 + nlo;
      #pragma unroll
      for (int i = 0; i < 8; ++i) {
        int r = rowbase + mt * 16 + khalf * 8 + i;
        og[(size_t)r * NCOL_PAD + c] = acc[mt][t][i];
      }
    }
}

// ---------------------------------------------------------------------------
// Post-process: softmax/exp, masked cumsum gates, outer product, max-normalize
// ---------------------------------------------------------------------------
__device__ __forceinline__ void axis_mask(const float* a, int lo, int hi,
                                          float* outm) {
  float s1 = 0.0f, s2 = 0.0f;
  #pragma unroll
  for (int w = 0; w < WD; ++w) {
    if (w >= lo) s1 += a[w];
    if (w < hi)  s2 += a[w];
  }
  float r1 = 1.0f / s1, r2 = 1.0f / s2;
  float c1 = 0.0f, cs2 = 0.0f;
  #pragma unroll
  for (int w = 0; w < WD; ++w) {
    c1  += (w >= lo ? a[w] : 0.0f) * r1;
    float t1 = (c1 >= 0.3f) ? c1 : 0.0f;
    cs2 += (w < hi ? a[w] : 0.0f) * r2;
    float c2 = 1.0f - cs2;
    float t2 = (c2 >= 0.3f) ? c2 : 0.0f;
    outm[w] = t1 * t2;
  }
}

__global__ __launch_bounds__(256) void post_kernel(
    const float* __restrict__ vec, const float* __restrict__ keypoint,
    float* __restrict__ out)
{
  int idx = blockIdx.x * blockDim.x + threadIdx.x;
  if (idx >= BSZ * 5) return;
  int p = idx / BSZ;          // part: head, arm, upper, lower, foot
  int b = idx - p * BSZ;      // sample

  gcf vg = (gcf)vec;
  gcf kg = (gcf)keypoint;
  gf  og = (gf)out;

  const int G  = (p == 4) ? 1 : 2;
  const int nk = (p == 4) ? 2 : 4;
  const int kidx_tab[5][4] = {{0,1,3,4},{7,8,9,10},{5,6,11,12},
                              {11,12,13,14},{15,16,15,16}};
  const int    coloff_tab[5] = {0, 28, 56, 84, 112};
  const size_t obase_tab[5]  = {0, 401408, 802816, 1204224, 1605632};

  int minx = 7, maxx = -8, miny = 7, maxy = -8;
  for (int j = 0; j < nk; ++j) {
    int ki = kidx_tab[p][j];
    float fx = kg[((size_t)b * 17 + ki) * 2 + 0];
    float fy = kg[((size_t)b * 17 + ki) * 2 + 1];
    int kx = (int)floorf(fx * 7.0f); if (kx > 6) kx = 6;
    int ky = (int)floorf(fy * 7.0f); if (ky > 6) ky = 6;
    minx = min(minx, kx); maxx = max(maxx, kx);
    miny = min(miny, ky); maxy = max(maxy, ky);
  }
  int lox = max(minx - 1, 0), hix = min(maxx + 1, WD - 1);
  int loy = max(miny - 1, 0), hiy = min(maxy + 1, WD - 1);

  float m[2][49];
  float mx = 0.0f;
  for (int g = 0; g < G; ++g) {
    gcf gv = vg + (size_t)b * NCOL_PAD + coloff_tab[p] + g * 14;
    float v0[WD], v1[WD];
    #pragma unroll
    for (int w = 0; w < WD; ++w) {
      float a0 = gv[w], a1 = gv[WD + w];
      float hi = fmaxf(a0, a1);
      float e0 = __expf(a0 - hi), e1 = __expf(a1 - hi);
      float s  = e0 + e1;
      v0[w] = __expf(e0 / s);   // exp(softmax) along the size-2 axis
      v1[w] = __expf(e1 / s);
    }
    float xm[WD], ym[WD];
    axis_mask(v0, lox, hix, xm);
    axis_mask(v1, loy, hiy, ym);
    for (int y = 0; y < WD; ++y)
      for (int xx = 0; xx < WD; ++xx) {
        float val = ym[y] * xm[xx];
        m[g][y * WD + xx] = val;
        mx = fmaxf(mx, val);
      }
  }
  float inv = 1.0f / (mx + 1e-7f);
  for (int g = 0; g < G; ++g) {
    gf o = og + obase_tab[p] + ((size_t)b * G + g) * 49;
    for (int i = 0; i < 49; ++i) o[i] = m[g][i] * inv;
  }
}

// ---------------------------------------------------------------------------
extern "C" void kernel_launch(void* const* d_in, const int* in_sizes, int n_in,
                              void* d_out, int out_size, void* d_ws, size_t ws_size,
                              hipStream_t stream) {
  (void)in_sizes; (void)n_in; (void)out_size; (void)ws_size;
  const float* x  = (const float*)d_in[0];
  const float* kp = (const float*)d_in[1];
  const float* Wh = (const float*)d_in[2];
  const float* bh = (const float*)d_in[3];
  const float* Wa = (const float*)d_in[4];
  const float* ba = (const float*)d_in[5];
  const float* Wu = (const float*)d_in[6];
  const float* bu = (const float*)d_in[7];
  const float* Wl = (const float*)d_in[8];
  const float* bl = (const float*)d_in[9];
  const float* Wf = (const float*)d_in[10];
  const float* bf = (const float*)d_in[11];
  float* vec = (float*)d_ws;          // 4096 x 128 f32 = 2 MB scratch
  float* out = (float*)d_out;

  gemm_kernel<<<BSZ / BROWS, 128, 0, stream>>>(x, Wh, bh, Wa, ba, Wu, bu,
                                               Wl, bl, Wf, bf, vec);
  post_kernel<<<(BSZ * 5 + 255) / 256, 256, 0, stream>>>(vec, kp, out);
}